// PredictiveCodingLayer_22497038696512
// MI455X (gfx1250) — compile-verified
//
#include <hip/hip_runtime.h>
#include <cstdint>
#include <cstddef>

#define N_TOK 8192
#define DDIM 1024
#define NEXP 8
#define CAP 8448            // 8192 + 8*16 padding, rounded up
#define MAXTILES 520        // 8192/16 + 8
#define LSTR 68             // LDS row stride (floats): 68%64==4 -> conflict-free

typedef float v2f __attribute__((ext_vector_type(2)));
typedef float v8f __attribute__((ext_vector_type(8)));
typedef int v4i __attribute__((ext_vector_type(4)));

#define AS1 __attribute__((address_space(1)))
#define AS3 __attribute__((address_space(3)))

__device__ __forceinline__ void async_b128(const float* gsrc, float* ldst) {
  __builtin_amdgcn_global_load_async_to_lds_b128(
      (AS1 v4i*)gsrc, (AS3 v4i*)ldst, 0, 0);
}

// ---------------- init: clear counters, mark bucket padding ----------------
__global__ __launch_bounds__(256) void init_kernel(int* __restrict__ bTok,
                                                   int* __restrict__ cnt) {
  int i = blockIdx.x * 256 + threadIdx.x;
  if (i < 2 * CAP) bTok[i] = -1;
  if (i < 2 * NEXP) cnt[i] = 0;
}

// ---------------- gating: logits, top-2, softmax, bucket counts ------------
__global__ __launch_bounds__(256) void gate_kernel(
    const float* __restrict__ mu, const float* __restrict__ gate_w,
    const float* __restrict__ gate_b, int2* __restrict__ topIdx,
    float2* __restrict__ topW, int* __restrict__ cnt) {
  __shared__ float gw[NEXP * DDIM];
  int tid = threadIdx.x;
  for (int i = tid; i < NEXP * DDIM; i += 256) gw[i] = gate_w[i];
  __syncthreads();

  int lane = tid & 31;
  int n = blockIdx.x * 8 + (tid >> 5);
  const float* mrow = mu + (size_t)n * DDIM;

  float acc[NEXP];
#pragma unroll
  for (int e = 0; e < NEXP; ++e) acc[e] = 0.f;
  for (int j = 0; j < DDIM / 32; ++j) {
    int d = j * 32 + lane;
    float m = mrow[d];
#pragma unroll
    for (int e = 0; e < NEXP; ++e) acc[e] = fmaf(m, gw[e * DDIM + d], acc[e]);
  }
#pragma unroll
  for (int e = 0; e < NEXP; ++e) {
#pragma unroll
    for (int off = 16; off > 0; off >>= 1)
      acc[e] += __shfl_xor(acc[e], off, 32);
  }
  if (lane == 0) {
#pragma unroll
    for (int e = 0; e < NEXP; ++e) acc[e] += gate_b[e];
    int i0 = 0;
    float l0 = acc[0];
#pragma unroll
    for (int e = 1; e < NEXP; ++e)
      if (acc[e] > l0) { l0 = acc[e]; i0 = e; }
    int i1 = -1;
    float l1 = -3.4e38f;
#pragma unroll
    for (int e = 0; e < NEXP; ++e)
      if (e != i0 && acc[e] > l1) { l1 = acc[e]; i1 = e; }
    float z = __expf(l1 - l0);            // <= 1
    float inv = 1.f / (1.f + z);
    topIdx[n] = make_int2(i0, i1);
    topW[n] = make_float2(inv, z * inv);
    atomicAdd(&cnt[i0], 1);
    atomicAdd(&cnt[NEXP + i1], 1);
  }
}

// ---------------- scan: bucket offsets + tile descriptors ------------------
__global__ void scan_kernel(const int* __restrict__ cnt, int* __restrict__ starts,
                            int* __restrict__ fill, int* __restrict__ tileExpert,
                            int* __restrict__ tileRow, int* __restrict__ numTiles) {
  if (threadIdx.x == 0 && blockIdx.x == 0) {
    for (int s = 0; s < 2; ++s) {
      int off = 0, nt = 0;
      for (int e = 0; e < NEXP; ++e) {
        starts[s * NEXP + e] = off;
        fill[s * NEXP + e] = 0;
        int c = cnt[s * NEXP + e];
        int tiles = (c + 15) >> 4;
        for (int t = 0; t < tiles; ++t) {
          tileExpert[s * MAXTILES + nt] = e;
          tileRow[s * MAXTILES + nt] = off + t * 16;
          ++nt;
        }
        off += tiles * 16;
      }
      numTiles[s] = nt;
    }
  }
}

// ---------------- scatter tokens into (slot, expert) buckets ---------------
__global__ __launch_bounds__(256) void scatter_kernel(
    const int2* __restrict__ topIdx, const float2* __restrict__ topW,
    const int* __restrict__ starts, int* __restrict__ fill,
    int* __restrict__ bTok, float* __restrict__ bW) {
  int n = blockIdx.x * 256 + threadIdx.x;
  if (n >= N_TOK) return;
  int2 idx = topIdx[n];
  float2 w = topW[n];
  int p0 = starts[idx.x] + atomicAdd(&fill[idx.x], 1);
  bTok[p0] = n;
  bW[p0] = w.x;
  int p1 = starts[NEXP + idx.y] + atomicAdd(&fill[NEXP + idx.y], 1);
  bTok[CAP + p1] = n;
  bW[CAP + p1] = w.y;
}

// ---------------- gathered expert GEMM via V_WMMA_F32_16X16X4_F32 ----------
// Tile: 16 tokens x 128 outputs per workgroup (8 waves x one 16x16 C tile).
// Global->LDS staging uses GLOBAL_LOAD_ASYNC_TO_LDS_B128 (ASYNCcnt), double-
// buffered. 9 async B128 per thread per K-chunk; in-order completion means
// s_wait_asynccnt<=9 after issuing chunk k+1 guarantees chunk k has landed.
__global__ __launch_bounds__(256) void gemm_kernel(
    const float* __restrict__ mu, const float* __restrict__ expert_w,
    const float* __restrict__ expert_b, const float* __restrict__ bias,
    const int* __restrict__ bTok, const float* __restrict__ bW,
    const int* __restrict__ tileExpert, const int* __restrict__ tileRow,
    const int* __restrict__ numTiles, float* __restrict__ out, int addMode) {
  __shared__ float As[2 * 16 * LSTR];    // 16 tokens x 64 K, double-buffered
  __shared__ float Bs[2 * 128 * LSTR];   // 128 outputs x 64 K, double-buffered
  __shared__ int sTok[16];
  __shared__ float sW[16];

  int t = blockIdx.y;
  if (t >= numTiles[0]) return;
  int e = tileExpert[t];
  int rbase = tileRow[t];
  int tid = threadIdx.x;
  int lane = tid & 31;
  int wv = tid >> 5;
  int i0 = blockIdx.x * 128;

  if (tid < 16) {
    sTok[tid] = bTok[rbase + tid];
    sW[tid] = bW[rbase + tid];
  }
  __syncthreads();

  int r16 = tid >> 4;  // 0..15: row within tile (A) / row group (B)
  int seg = tid & 15;  // 0..15: 4-float segment of the 64-K chunk

  int arow = sTok[r16] < 0 ? 0 : sTok[r16];
  const float* aPtr = mu + (size_t)arow * DDIM + seg * 4;
  const float* bPtr =
      expert_w + (size_t)e * DDIM * DDIM + (size_t)(i0 + r16) * DDIM + seg * 4;
  float* aDst = &As[r16 * LSTR + seg * 4];
  float* bDst = &Bs[r16 * LSTR + seg * 4];

  const int NCHUNK = DDIM / 64;

  // prologue: async-load chunk 0 into buffer 0
  async_b128(aPtr, aDst);
#pragma unroll
  for (int j = 0; j < 8; ++j)
    async_b128(bPtr + (size_t)j * 16 * DDIM, bDst + j * 16 * LSTR);
  __builtin_amdgcn_s_wait_asynccnt(0);
  __syncthreads();

  v8f acc = {};
  int m = lane & 15;
  int h = lane >> 4;
  int buf = 0;
  for (int kc = 0; kc < NCHUNK; ++kc) {
    if (kc + 1 < NCHUNK) {
      // all waves are done reading buf^1 (they read it as chunk kc-1)
      __syncthreads();
      int d0 = (kc + 1) * 64;
      int nb = buf ^ 1;
      async_b128(aPtr + d0, aDst + nb * 16 * LSTR);
#pragma unroll
      for (int j = 0; j < 8; ++j)
        async_b128(bPtr + (size_t)j * 16 * DDIM + d0,
                   bDst + nb * 128 * LSTR + j * 16 * LSTR);
      // in-order completion: <=9 outstanding => chunk kc fully in LDS
      __builtin_amdgcn_s_wait_asynccnt(9);
      __syncthreads();
    }
    // WMMA operand layout: A[v][lane]=A(M=lane%16, K=v+2*(lane/16))
    //                      B[v][lane]=B(K=v+2*(lane/16), N=lane%16)
    const float* Ab = &As[buf * 16 * LSTR + m * LSTR + 2 * h];
    const float* Bb = &Bs[buf * 128 * LSTR + (wv * 16 + m) * LSTR + 2 * h];
#pragma unroll
    for (int kk = 0; kk < 16; ++kk) {
      v2f a = *(const v2f*)(Ab + kk * 4);
      v2f b = *(const v2f*)(Bb + kk * 4);
      acc = __builtin_amdgcn_wmma_f32_16x16x4_f32(false, a, false, b,
                                                  (short)0, acc, false, false);
    }
    buf ^= 1;
  }

  // epilogue: C VGPR r holds (M=r+8h, N=lane%16)
  int col = i0 + wv * 16 + m;
  float eb = expert_b[e * DDIM + col];
  float bc = bias[col];
#pragma unroll
  for (int r = 0; r < 8; ++r) {
    int rr = r + 8 * h;
    int tok = sTok[rr];
    if (tok >= 0) {
      float val = sW[rr] * (acc[r] + eb);
      float* p = out + (size_t)tok * DDIM + col;
      if (addMode)
        *p = *p + val;       // slot-1 pass: accumulate (ordered by launch)
      else
        *p = val + bc;       // slot-0 pass: initialize with bias
    }
  }
}

// ---------------------------------------------------------------------------
extern "C" void kernel_launch(void* const* d_in, const int* in_sizes, int n_in,
                              void* d_out, int out_size, void* d_ws,
                              size_t ws_size, hipStream_t stream) {
  (void)in_sizes; (void)n_in; (void)out_size; (void)ws_size;
  const float* mu = (const float*)d_in[0];
  const float* gate_w = (const float*)d_in[1];
  const float* gate_b = (const float*)d_in[2];
  const float* expert_w = (const float*)d_in[3];
  const float* expert_b = (const float*)d_in[4];
  const float* bias = (const float*)d_in[5];
  float* out = (float*)d_out;

  char* w = (char*)d_ws;
  int2* topIdx = (int2*)w;            w += (size_t)N_TOK * sizeof(int2);
  float2* topW = (float2*)w;          w += (size_t)N_TOK * sizeof(float2);
  int* cnt = (int*)w;                 w += 256;
  int* starts = (int*)w;              w += 256;
  int* fill = (int*)w;                w += 256;
  int* numTiles = (int*)w;            w += 256;
  int* tileExpert = (int*)w;          w += 2 * MAXTILES * sizeof(int) + 64;
  int* tileRow = (int*)w;             w += 2 * MAXTILES * sizeof(int) + 64;
  int* bTok = (int*)w;                w += 2 * CAP * sizeof(int);
  float* bW = (float*)w;              w += 2 * CAP * sizeof(float);

  init_kernel<<<(2 * CAP + 255) / 256, 256, 0, stream>>>(bTok, cnt);
  gate_kernel<<<N_TOK / 8, 256, 0, stream>>>(mu, gate_w, gate_b, topIdx, topW,
                                             cnt);
  scan_kernel<<<1, 32, 0, stream>>>(cnt, starts, fill, tileExpert, tileRow,
                                    numTiles);
  scatter_kernel<<<N_TOK / 256, 256, 0, stream>>>(topIdx, topW, starts, fill,
                                                  bTok, bW);
  dim3 grid(DDIM / 128, MAXTILES);
  // slot 0 (top-1 experts): out = bias + w0*(mu·We^T + eb)
  gemm_kernel<<<grid, 256, 0, stream>>>(mu, expert_w, expert_b, bias, bTok, bW,
                                        tileExpert, tileRow, numTiles, out, 0);
  // slot 1 (top-2 experts): out += w1*(mu·We^T + eb)
  gemm_kernel<<<grid, 256, 0, stream>>>(mu, expert_w, expert_b, bias,
                                        bTok + CAP, bW + CAP,
                                        tileExpert + MAXTILES,
                                        tileRow + MAXTILES, numTiles + 1, out,
                                        1);
}